// NMSLayer_20169166422186
// MI455X (gfx1250) — compile-verified
//
#include <hip/hip_runtime.h>
#include <hip/hip_bf16.h>
#include <math.h>

#define THREADS 256

typedef __attribute__((ext_vector_type(2))) float v2f;
typedef __attribute__((ext_vector_type(8))) float v8f;
typedef __attribute__((ext_vector_type(4))) unsigned int u32x4;
typedef __attribute__((ext_vector_type(8))) int i32x8;
typedef __attribute__((ext_vector_type(4))) int i32x4;

#if __has_builtin(__builtin_amdgcn_tensor_load_to_lds)
#define HAVE_TDM 1
#else
#define HAVE_TDM 0
#endif

#if HAVE_TDM
// 1-D contiguous DMA: ndw dwords from global -> LDS via the Tensor Data Mover.
// D# per CDNA5 ISA: group0 = {count=1, lds_addr,
// global_addr[56:0], type=2}; group1 = {data_size=4B, tensor_dim0=tile_dim0=ndw,
// tensor_dim1=tile_dim1=1, stride0=ndw}; groups 2/3 = dims 2..4 == 1 (unused).
__device__ __forceinline__ void tdm_copy_1d(const void* gsrc, void* ldst, unsigned ndw) {
  unsigned long long ga = (unsigned long long)(uintptr_t)gsrc;
  unsigned la = (unsigned)(unsigned long long)(uintptr_t)ldst;  // low 32 bits of flat LDS addr == LDS offset
  u32x4 g0 = { 1u, la, (unsigned)ga,
               (unsigned)((ga >> 32) & 0x1ffffffull) | 0x80000000u };
  i32x8 g1 = { (int)(2u << 16),                               // data_size = 4B
               (int)((ndw & 0xffffu) << 16),                  // tensor_dim0[15:0]
               (int)(((ndw >> 16) & 0xffffu) | (1u << 16)),   // tensor_dim0[31:16], tensor_dim1=1
               (int)((ndw & 0xffffu) << 16),                  // tile_dim0
               1,                                             // tile_dim1=1, tile_dim2=0
               (int)ndw,                                      // tensor_dim0_stride[31:0]
               0, 0 };
  i32x4 g2 = { 1, 1, 0, 0 };                                  // tensor_dim2=1, tensor_dim3=1
  i32x4 g3 = { 0, (int)(1u << 16), 0, 0 };                    // tensor_dim4=1
#if defined(__clang_major__) && (__clang_major__ >= 23)
  i32x8 g4 = { 0, 0, 0, 0, 0, 0, 0, 0 };
  __builtin_amdgcn_tensor_load_to_lds(g0, g1, g2, g3, g4, 0);
#else
  __builtin_amdgcn_tensor_load_to_lds(g0, g1, g2, g3, 0);
#endif
#if __has_builtin(__builtin_amdgcn_s_wait_tensorcnt)
  __builtin_amdgcn_s_wait_tensorcnt(0);
#else
  asm volatile("s_wait_tensorcnt 0x0" ::: "memory");
#endif
}
#endif

__device__ __forceinline__ unsigned wave_ballot32(bool p) {
#if __has_builtin(__builtin_amdgcn_ballot_w32)
  return __builtin_amdgcn_ballot_w32(p);
#else
  return (unsigned)__ballot(p);
#endif
}

// ---------------------------------------------------------------------------
// K1: deinterleave x[B,N,84] -> boxes[B,N,4] and scores_t[B,C,N] (transposed
// through LDS so both reads and writes are coalesced).
// ---------------------------------------------------------------------------
__global__ __launch_bounds__(THREADS) void k_prep(const float* __restrict__ x,
                                                  float* __restrict__ boxes,
                                                  float* __restrict__ scores_t) {
  __shared__ float tile[64 * 84];
  const int b = blockIdx.y, n0 = blockIdx.x * 64, tid = threadIdx.x;
  const float* src = x + ((size_t)b * 2048 + n0) * 84;
  for (int k = tid; k < 64 * 84; k += THREADS) tile[k] = src[k];
  __syncthreads();
  {  // 64 rows x 4 comps = 256 = blockDim
    int r = tid >> 2, cmp = tid & 3;
    boxes[((size_t)b * 2048 + n0 + r) * 4 + cmp] = tile[r * 84 + cmp];
  }
  for (int k = tid; k < 64 * 80; k += THREADS) {
    int c = k >> 6, r = k & 63;
    scores_t[((size_t)b * 80 + c) * 2048 + n0 + r] = tile[r * 84 + 4 + c];
  }
}

// ---------------------------------------------------------------------------
// K2: per-batch (iou > 0.5) bit mask, mask[b][i] = 64 u32 words over j.
// Boxes TDM-DMA'd to LDS. Each wave owns 16 rows; per 16-col tile the
// union = area_i + area_j - inter is computed by v_wmma_f32_16x16x4_f32
// (A=[area_row,1], B=[1;area_col], C=-inter) co-executing with the VALU
// intersection math. iou>0.5  <=>  inter > 0.5*union (division-free).
// Bits packed via ballot per C/D accumulator VGPR (rows v and v+8).
// ---------------------------------------------------------------------------
__global__ __launch_bounds__(THREADS) void k_iou_mask(const float* __restrict__ boxes,
                                                      unsigned* __restrict__ mask) {
  __shared__ float4 bx[2048];
  __shared__ float ar[2048];
  const int b = blockIdx.y, tid = threadIdx.x;
  const int lane = tid & 31, w = tid >> 5;
#if HAVE_TDM
  if (w == 0) tdm_copy_1d(boxes + (size_t)b * 2048 * 4, (void*)bx, 2048 * 4);
#else
  for (int k = tid; k < 2048 * 4; k += THREADS)
    ((float*)bx)[k] = boxes[(size_t)b * 2048 * 4 + k];
#endif
  __syncthreads();
  for (int k = tid; k < 2048; k += THREADS) {
    float4 v = bx[k];
    ar[k] = (v.z - v.x) * (v.w - v.y);
  }
  __syncthreads();

  const int r0 = blockIdx.x * 128 + w * 16;   // 16 rows per wave
  const int mofs = (lane >= 16) ? 8 : 0;      // D-matrix M offset for this half-wave
  float4 rbox[8];
#pragma unroll
  for (int v = 0; v < 8; ++v) rbox[v] = bx[r0 + v + mofs];
  v2f A;
  A[0] = (lane < 16) ? ar[r0 + lane] : 0.f;   // K=0: area of row M=lane
  A[1] = (lane < 16) ? 1.f : 0.f;             // K=1: ones ; K=2,3 (hi lanes) = 0

  unsigned row_word = 0u;
  for (int ct = 0; ct < 128; ++ct) {
    const int col = ct * 16 + (lane & 15);
    const float4 cb = bx[col];
    v2f B;
    B[0] = (lane < 16) ? 1.f : 0.f;           // K=0 row of B: ones
    B[1] = (lane < 16) ? ar[col] : 0.f;       // K=1 row of B: area of col N
    float inter[8];
    v8f cm;
#pragma unroll
    for (int v = 0; v < 8; ++v) {
      float4 rb = rbox[v];
      float ix = fminf(rb.z, cb.z) - fmaxf(rb.x, cb.x);
      float iy = fminf(rb.w, cb.w) - fmaxf(rb.y, cb.y);
      float it = fmaxf(ix, 0.f) * fmaxf(iy, 0.f);
      inter[v] = it;
      cm[v] = -it;
    }
#if __has_builtin(__builtin_amdgcn_wmma_f32_16x16x4_f32)
    v8f uni = __builtin_amdgcn_wmma_f32_16x16x4_f32(false, A, false, B,
                                                    (short)0, cm, false, false);
#else
    v8f uni;
#pragma unroll
    for (int v = 0; v < 8; ++v)
      uni[v] = ar[r0 + v + mofs] + ar[col] - inter[v];
#endif
    const unsigned shift = (ct & 1) ? 16u : 0u;
#pragma unroll
    for (int v = 0; v < 8; ++v) {
      bool p = inter[v] > 0.5f * uni[v];
      unsigned bal = wave_ballot32(p);   // bits 0..15: row v; bits 16..31: row v+8
      unsigned contrib = (lane == v) ? (bal & 0xffffu)
                        : ((lane == v + 8) ? (bal >> 16) : 0u);
      row_word |= contrib << shift;
    }
    if (ct & 1) {
      if (lane < 16)
        mask[((size_t)b * 2048 + r0 + lane) * 64 + (ct >> 1)] = row_word;
      row_word = 0u;
    }
  }
}

// ---------------------------------------------------------------------------
// K3: one workgroup per (b,c). TDM-DMA the class score row to LDS, build
// u64 keys (score_bits<<32 | (2047-idx) for stable tie order), bitonic sort
// descending in LDS, then greedy NMS using the precomputed bit mask
// (64-word row fetched into LDS per kept box). Emits <=200 sorted
// candidates per class, padded with -inf.
// ---------------------------------------------------------------------------
__global__ __launch_bounds__(THREADS) void k_nms(const float* __restrict__ scores_t,
                                                 const unsigned* __restrict__ mask,
                                                 float* __restrict__ cand_score,
                                                 unsigned* __restrict__ cand_meta) {
  __shared__ unsigned long long keys[2048];
  __shared__ float sc[2048];
  __shared__ unsigned rowbuf[64];
  __shared__ unsigned char sup[2048];
  const int c = blockIdx.x, b = blockIdx.y, tid = threadIdx.x;
  const float* ssrc = scores_t + ((size_t)b * 80 + c) * 2048;
#if HAVE_TDM
  if ((tid >> 5) == 0) tdm_copy_1d(ssrc, (void*)sc, 2048);
#else
  for (int k = tid; k < 2048; k += THREADS) sc[k] = ssrc[k];
#endif
  __syncthreads();
  for (int k = tid; k < 2048; k += THREADS) {
    unsigned sb = __float_as_uint(sc[k]);  // scores >= 0 -> bits are order-preserving
    keys[k] = ((unsigned long long)sb << 32) | (unsigned)(2047 - k);
    sup[k] = 0;
  }
  __syncthreads();
  // bitonic sort, descending
  for (unsigned kk = 2; kk <= 2048; kk <<= 1) {
    for (unsigned j = kk >> 1; j > 0; j >>= 1) {
      for (unsigned i = tid; i < 2048; i += THREADS) {
        unsigned l = i ^ j;
        if (l > i) {
          unsigned long long a = keys[i], d = keys[l];
          bool up = ((i & kk) == 0);
          if (up ? (a < d) : (a > d)) { keys[i] = d; keys[l] = a; }
        }
      }
      __syncthreads();
    }
  }
  int kcount = 0;
  const size_t cbase = ((size_t)b * 80 + c) * 200;
  for (int i = 0; i < 2048; ++i) {
    unsigned long long ki = keys[i];
    float si = __uint_as_float((unsigned)(ki >> 32));
    if (!(si > 0.01f)) break;          // sorted: everything after also fails
    if (sup[i]) continue;
    unsigned oi = 2047u - (unsigned)(ki & 0xffffffffull);
    if (tid == 0) {
      cand_score[cbase + kcount] = si;
      cand_meta[cbase + kcount] = ((unsigned)c << 11) | oi;
    }
    ++kcount;
    if (kcount >= 200) break;          // rank cap == OUTPUT_SIZE
    const unsigned* mrow = mask + ((size_t)b * 2048 + oi) * 64;
    if (tid < 64) rowbuf[tid] = mrow[tid];
    __syncthreads();
    for (int jj = i + 1 + tid; jj < 2048; jj += THREADS) {
      unsigned oj = 2047u - (unsigned)(keys[jj] & 0xffffffffull);
      if ((rowbuf[oj >> 5] >> (oj & 31u)) & 1u) sup[jj] = 1;
    }
    __syncthreads();
  }
  __syncthreads();
  for (int s2 = kcount + tid; s2 < 200; s2 += THREADS) {
    cand_score[cbase + s2] = -INFINITY;
    cand_meta[cbase + s2] = 0u;
  }
}

// ---------------------------------------------------------------------------
// K4: per-batch 80-way merge of sorted class lists -> global top-200 rows
// [cls, score, x1,y1,x2,y2] (boxes clipped to [0,1]); -inf => zero row.
// Tie-break = lower slot at each reduction step (not always the lowest class).
// ---------------------------------------------------------------------------
__global__ __launch_bounds__(128) void k_merge(const float* __restrict__ cand_score,
                                               const unsigned* __restrict__ cand_meta,
                                               const float* __restrict__ boxes,
                                               float* __restrict__ out) {
  __shared__ float hv[128];
  __shared__ int hidx[128];
  __shared__ int ptr[80];
  const int b = blockIdx.x, tid = threadIdx.x;
  if (tid < 80) ptr[tid] = 0;
  __syncthreads();
  for (int round = 0; round < 200; ++round) {
    float v = -INFINITY;
    if (tid < 80) {
      int p = ptr[tid];
      if (p < 200) v = cand_score[((size_t)b * 80 + tid) * 200 + p];
    }
    hv[tid] = v; hidx[tid] = tid;
    __syncthreads();
    for (int s = 64; s > 0; s >>= 1) {
      if (tid < s) {
        if (hv[tid + s] > hv[tid]) { hv[tid] = hv[tid + s]; hidx[tid] = hidx[tid + s]; }
      }
      __syncthreads();
    }
    float best = hv[0];
    int bc = hidx[0];
    int p = (best != -INFINITY) ? ptr[bc] : 0;
    __syncthreads();
    if (tid < 6) {
      float val = 0.f;
      if (best != -INFINITY) {
        if (tid == 0) val = (float)bc;
        else if (tid == 1) val = best;
        else {
          unsigned meta = cand_meta[((size_t)b * 80 + bc) * 200 + p];
          unsigned oi = meta & 2047u;
          float bv = boxes[((size_t)b * 2048 + oi) * 4 + (tid - 2)];
          val = fminf(fmaxf(bv, 0.f), 1.f);
        }
      }
      out[((size_t)b * 200 + round) * 6 + tid] = val;
    }
    if (tid == 0 && best != -INFINITY) ptr[bc] = p + 1;
    __syncthreads();
  }
}

extern "C" void kernel_launch(void* const* d_in, const int* in_sizes, int n_in,
                              void* d_out, int out_size, void* d_ws, size_t ws_size,
                              hipStream_t stream) {
  (void)in_sizes; (void)n_in; (void)out_size; (void)ws_size;
  const float* x = (const float*)d_in[0];
  float* out = (float*)d_out;

  // Workspace layout (elements):
  //   boxes      f32  [8*2048*4]   =   65536
  //   scores_t   f32  [8*80*2048]  = 1310720
  //   cand_score f32  [8*80*200]   =  128000
  //   cand_meta  u32  [8*80*200]   =  128000
  //   mask       u32  [8*2048*64]  = 1048576     (~10.7 MB total)
  float* fws = (float*)d_ws;
  float* boxes = fws;
  float* scores_t = fws + 65536;
  float* cand_score = fws + 65536 + 1310720;
  unsigned* cand_meta = (unsigned*)d_ws + 65536 + 1310720 + 128000;
  unsigned* mask = (unsigned*)d_ws + 65536 + 1310720 + 128000 + 128000;

  k_prep<<<dim3(32, 8), THREADS, 0, stream>>>(x, boxes, scores_t);
  k_iou_mask<<<dim3(16, 8), THREADS, 0, stream>>>(boxes, mask);
  k_nms<<<dim3(80, 8), THREADS, 0, stream>>>(scores_t, mask, cand_score, cand_meta);
  k_merge<<<8, 128, 0, stream>>>(cand_score, cand_meta, boxes, out);
}